// GCN_58600533787398
// MI455X (gfx1250) — compile-verified
//
#include <hip/hip_runtime.h>
#include <hip/hip_bf16.h>
#include <stdint.h>

typedef __attribute__((ext_vector_type(2))) float v2f;
typedef __attribute__((ext_vector_type(4))) float v4f;
typedef __attribute__((ext_vector_type(8))) float v8f;
typedef __attribute__((ext_vector_type(4))) unsigned int u32x4;
typedef __attribute__((ext_vector_type(4))) int i32x4;
typedef __attribute__((ext_vector_type(8))) int i32x8;

#define D_DIM   128
#define WSTR    132          // LDS row stride in floats: 132 % 64 == 4 -> conflict-free b64 reads
#define WAVES   4            // waves per GEMM block

#if defined(__gfx1250__) && __has_builtin(__builtin_amdgcn_tensor_load_to_lds)
#define USE_TDM 1
#else
#define USE_TDM 0
#endif

// ---------------------------------------------------------------------------
// hardware fp32 atomic add (no-return form -> STOREcnt, no CAS loop)
// ---------------------------------------------------------------------------
__device__ __forceinline__ void hw_atomic_add_f32(float* addr, float val) {
    asm volatile("global_atomic_add_f32 %0, %1, off"
                 :
                 : "v"(addr), "v"(val)
                 : "memory");
}

#if USE_TDM
// ---------------------------------------------------------------------------
// TDM 2-D tile load Global->LDS with hardware row padding (ISA 08 sect. 8).
//   - 128-element (f32) rows, `rows` of them, global row stride = 128 elems
//   - pad_interval code 6 (pad after every 128 DWORDs stored),
//     pad_amount  code 3 (insert 4 DWORDs)  => LDS row stride = 132 floats
// 6-arg builtin form (amdgpu-toolchain / clang-23):
//   (u32x4 g0, i32x8 g1, i32x4, i32x4, i32x8, i32 cpol)
// ---------------------------------------------------------------------------
__device__ __forceinline__ void tdm_load_rows_padded(uint32_t lds_byte_off,
                                                     const void* gaddr,
                                                     uint32_t rows) {
    const uint64_t ga = (uint64_t)(uintptr_t)gaddr;
    u32x4 g0;
    g0.x = 1u;                                                  // count=1, user D#
    g0.y = lds_byte_off;                                        // lds_addr (bytes)
    g0.z = (uint32_t)ga;                                        // global_addr[31:0]
    g0.w = ((uint32_t)(ga >> 32) & 0x01FFFFFFu) | (2u << 30);   // addr[56:32] | type=2
    i32x8 g1;
    g1[0] = (2 << 16)        // data_size = 4 bytes
          | (1 << 20)        // pad_enable
          | (6 << 22)        // pad_interval: every 128 DWORDs
          | (3 << 25);       // pad_amount: 4 DWORDs
    g1[1] = (int)(D_DIM << 16);     // tensor_dim0[15:0] = 128   (in hi half)
    g1[2] = (int)(rows  << 16);     // tensor_dim0[31:16]=0 | tensor_dim1[15:0]=rows
    g1[3] = (int)(D_DIM << 16);     // tensor_dim1[31:16]=0 | tile_dim0 = 128
    g1[4] = (int)rows;              // tile_dim1 = rows, tile_dim2 = 0 (2-D)
    g1[5] = D_DIM;                  // tensor_dim0_stride = 128 elems
    g1[6] = 0;                      // stride hi bits / dim1_stride (unused, 2-D)
    g1[7] = 0;
    const i32x4 z4 = {0, 0, 0, 0};                  // groups 2/3: dims 2..4 unused
    const i32x8 z8 = {0, 0, 0, 0, 0, 0, 0, 0};
    __builtin_amdgcn_tensor_load_to_lds(g0, g1, z4, z4, z8, 0);
}
#endif

// ---------------------------------------------------------------------------
// zero the aggregation buffer (workspace is poisoned, must clear every call)
// ---------------------------------------------------------------------------
__global__ void gcn_zero_kernel(v4f* __restrict__ p, int n4) {
    int i = blockIdx.x * blockDim.x + threadIdx.x;
    if (i < n4) {
        v4f z = {0.f, 0.f, 0.f, 0.f};
        p[i] = z;
    }
}

// ---------------------------------------------------------------------------
// SpMM scatter: one wave per edge; lane l handles floats [4l, 4l+4)
// agg[dst] += val * seq[src]
// ---------------------------------------------------------------------------
__global__ __launch_bounds__(256) void gcn_spmm_kernel(
    const float* __restrict__ seq,
    const int*   __restrict__ esrc,
    const int*   __restrict__ edst,
    const float* __restrict__ eval,
    float*       __restrict__ agg,
    int nE)
{
    const int wave = (blockIdx.x * blockDim.x + threadIdx.x) >> 5;
    const int lane = threadIdx.x & 31;
    if (wave >= nE) return;

    const int   s = esrc[wave];
    const int   d = edst[wave];
    const float v = eval[wave];

    const v4f x = *(const v4f*)(seq + (size_t)s * D_DIM + (lane << 2));
    float* dst = agg + (size_t)d * D_DIM + (lane << 2);
    hw_atomic_add_f32(dst + 0, x.x * v);
    hw_atomic_add_f32(dst + 1, x.y * v);
    hw_atomic_add_f32(dst + 2, x.z * v);
    hw_atomic_add_f32(dst + 3, x.w * v);
}

// ---------------------------------------------------------------------------
// Dense GEMM  y = agg @ W^T  with fused PReLU, using V_WMMA_F32_16X16X4_F32.
// Each wave computes a 16x128 output strip (one M-tile, eight 16x16 N-tiles).
//   A (16x4 f32):  lane L holds A[L&15][kb..kb+1], kb = 4j + 2*(L>>4)
//   B (4x16 f32):  lane L holds B[kb'..][L&15]  -> float2 W[o][kb..kb+1]
//   C/D (16x16):   vgpr v, lane L -> row v + 8*(L>>4), col L&15
// K loop is double-buffered: fragments for chunk j+1 are fetched from LDS
// while the 8 independent WMMAs of chunk j execute.
// ---------------------------------------------------------------------------
__global__ __launch_bounds__(WAVES * 32) void gcn_gemm_prelu_kernel(
    const float* __restrict__ agg,
    const float* __restrict__ W,
    const float* __restrict__ alpha_p,
    float*       __restrict__ out,
    int tilesM)
{
    extern __shared__ float smem[];
    float* Wl = smem;                      // 128 x WSTR
    float* Al = smem + D_DIM * WSTR;       // WAVES x 16 x WSTR

    const int tid  = threadIdx.x;
    const int lane = tid & 31;
    const int wid  = tid >> 5;

    int tile = blockIdx.x * WAVES + wid;
    if (tile >= tilesM) tile = tilesM - 1;     // clamp tail: EXEC stays all-1s

    float* Am = Al + wid * 16 * WSTR;
    const float* arow = agg + (size_t)tile * (16 * D_DIM);

#if USE_TDM
    // ---- Tensor Data Mover staging: W once (wave 0), A tile per wave ----
    if (wid == 0)
        tdm_load_rows_padded((uint32_t)(uintptr_t)(void*)Wl, W, D_DIM);
    tdm_load_rows_padded((uint32_t)(uintptr_t)(void*)Am, arow, 16);
    __builtin_amdgcn_s_wait_tensorcnt(0);
    __syncthreads();
#else
    // ---- manual staged copies (fallback) ----
#pragma unroll
    for (int it = 0; it < 32; ++it) {
        int idx = it * (WAVES * 32) + tid;          // 4096 float4 total
        int row = idx >> 5;
        int c   = (idx & 31) << 2;
        *(v4f*)(Wl + row * WSTR + c) = *(const v4f*)(W + (row << 7) + c);
    }
#pragma unroll
    for (int it = 0; it < 16; ++it)
        *(v4f*)(Am + it * WSTR + (lane << 2)) =
            *(const v4f*)(arow + (it << 7) + (lane << 2));
    __syncthreads();
#endif

    const int r  = lane & 15;          // matrix row within fragment
    const int h2 = (lane >> 4) << 1;   // K sub-offset: 0 (lanes 0-15) / 2 (16-31)

    v8f acc[8];
#pragma unroll
    for (int t = 0; t < 8; ++t) {
        v8f z = {0.f, 0.f, 0.f, 0.f, 0.f, 0.f, 0.f, 0.f};
        acc[t] = z;
    }

    const float* aBase = Am + r * WSTR + h2;
    const float* bBase = Wl + r * WSTR + h2;

    // prologue: fragments for chunk 0
    v2f a0 = *(const v2f*)(aBase);
    v2f b0[8];
#pragma unroll
    for (int t = 0; t < 8; ++t) b0[t] = *(const v2f*)(bBase + t * (16 * WSTR));

#pragma unroll 2
    for (int j = 0; j < 32; ++j) {
        const int kbn = (((j + 1) & 31) << 2);     // next chunk (wraps, harmless)
        v2f a1 = *(const v2f*)(aBase + kbn);
        v2f b1[8];
#pragma unroll
        for (int t = 0; t < 8; ++t)
            b1[t] = *(const v2f*)(bBase + t * (16 * WSTR) + kbn);
#pragma unroll
        for (int t = 0; t < 8; ++t)
            acc[t] = __builtin_amdgcn_wmma_f32_16x16x4_f32(
                false, a0, false, b0[t], (short)0, acc[t], false, false);
        a0 = a1;
#pragma unroll
        for (int t = 0; t < 8; ++t) b0[t] = b1[t];
    }

    // ---- PReLU epilogue + store ----
    const float alpha = *alpha_p;
    const int rowHalf = (lane >> 4) << 3;   // 0 or 8
#pragma unroll
    for (int t = 0; t < 8; ++t) {
#pragma unroll
        for (int v = 0; v < 8; ++v) {
            float y = acc[t][v];
            y = (y >= 0.f) ? y : alpha * y;
            out[(size_t)(tile * 16 + v + rowHalf) * D_DIM + t * 16 + r] = y;
        }
    }
}

// ---------------------------------------------------------------------------
extern "C" void kernel_launch(void* const* d_in, const int* in_sizes, int n_in,
                              void* d_out, int out_size, void* d_ws, size_t ws_size,
                              hipStream_t stream) {
    const float* seq   = (const float*)d_in[0];
    const int*   esrc  = (const int*)d_in[1];
    const int*   edst  = (const int*)d_in[2];
    const float* eval  = (const float*)d_in[3];
    const float* W     = (const float*)d_in[4];
    const float* alpha = (const float*)d_in[5];
    float*       out   = (float*)d_out;
    float*       agg   = (float*)d_ws;     // [nNodes][128] fp32 scratch

    const int nNodes = in_sizes[0] / D_DIM;   // 50000
    const int nE     = in_sizes[1];           // 800000

    // 1) clear aggregation buffer
    const int n4 = nNodes * (D_DIM / 4);
    gcn_zero_kernel<<<(n4 + 255) / 256, 256, 0, stream>>>((v4f*)agg, n4);

    // 2) edge scatter (8 edges per 256-thread block)
    gcn_spmm_kernel<<<(nE + 7) / 8, 256, 0, stream>>>(seq, esrc, edst, eval, agg, nE);

    // 3) dense GEMM + PReLU via f32 WMMA (TDM-staged LDS tiles)
    const int tilesM = nNodes / 16;           // 3125 (exact)
    const int grid   = (tilesM + WAVES - 1) / WAVES;
    const size_t shmem = (size_t)(D_DIM * WSTR + WAVES * 16 * WSTR) * sizeof(float);
    gcn_gemm_prelu_kernel<<<grid, WAVES * 32, shmem, stream>>>(agg, W, alpha, out, tilesM);
}